// InterpolativeUpsampler_25262997635079
// MI455X (gfx1250) — compile-verified
//
#include <hip/hip_runtime.h>

// Fused nearest-2x upsample + depthwise 3x3 [1,2,1]^2/16 blur.
// out[2i+a][2j+b] = sum over 2x2 input pixels with weights {1/4,3/4}x{1/4,3/4}.
// Memory-bound: 128 MiB read + 512 MiB NT-write => ~27us floor @ 23.3 TB/s.
// CDNA5 path: TDM tensor_load_to_lds stages input row tiles, s_wait_tensorcnt,
// compute from LDS, non-temporal B128 output stores.

#define HDIM 128
#define WDIM 128
#define TI   32      // input rows per block tile
#define W2   256     // output width

typedef float        f32x4 __attribute__((ext_vector_type(4)));
typedef unsigned int u32x4 __attribute__((ext_vector_type(4)));
typedef int          i32x4 __attribute__((ext_vector_type(4)));
typedef int          i32x8 __attribute__((ext_vector_type(8)));

#if defined(__HIP_DEVICE_COMPILE__) && __has_builtin(__builtin_amdgcn_tensor_load_to_lds)
#define UPS_HAVE_TDM 1
#else
#define UPS_HAVE_TDM 0
#endif

#if defined(__HIP_DEVICE_COMPILE__) && __has_builtin(__builtin_amdgcn_s_wait_tensorcnt)
#define UPS_WAIT_TENSOR() __builtin_amdgcn_s_wait_tensorcnt(0)
#elif defined(__HIP_DEVICE_COMPILE__)
#define UPS_WAIT_TENSOR() asm volatile("s_wait_tensorcnt 0" ::: "memory")
#else
#define UPS_WAIT_TENSOR()
#endif

__device__ __forceinline__ void hpass(float lf, f32x4 c, float rt, float h[8]) {
  h[0] = 0.25f * lf  + 0.75f * c.x;
  h[1] = 0.75f * c.x + 0.25f * c.y;
  h[2] = 0.25f * c.x + 0.75f * c.y;
  h[3] = 0.75f * c.y + 0.25f * c.z;
  h[4] = 0.25f * c.y + 0.75f * c.z;
  h[5] = 0.75f * c.z + 0.25f * c.w;
  h[6] = 0.25f * c.z + 0.75f * c.w;
  h[7] = 0.75f * c.w + 0.25f * rt;
}

#if UPS_HAVE_TDM
// Build a 2D TDM descriptor (CDNA5 ISA) and issue
// TENSOR_LOAD_TO_LDS: tile_rows x 128 f32 rows, row stride 128 elems.
__device__ __forceinline__ void tdm_load_2d(unsigned lds_byte_addr,
                                            const float* gsrc,
                                            unsigned tile_rows) {
  unsigned long long ga = (unsigned long long)(const void*)gsrc;
  u32x4 g0;
  g0.x = 1u;                                                    // count=1 (user desc)
  g0.y = lds_byte_addr;                                         // lds_addr [63:32]
  g0.z = (unsigned)(ga & 0xFFFFFFFFull);                        // global_addr[31:0]
  g0.w = (unsigned)((ga >> 32) & 0x01FFFFFFull) | (2u << 30);   // addr[56:32] | type=2

  const unsigned long long dim0    = (unsigned long long)WDIM;  // tensor_dim0
  const unsigned long long dim1    = (unsigned long long)HDIM;  // tensor_dim1
  const unsigned long long tdim0   = (unsigned long long)WDIM;  // tile_dim0
  const unsigned long long tdim1   = (unsigned long long)tile_rows;
  const unsigned long long stride0 = (unsigned long long)WDIM;  // dim0 stride (elems)

  unsigned long long q0 = (2ull << 16)                          // data_size = 4B
                        | ((dim0 & 0xFFFFull) << 48);           // tensor_dim0[15:0]
  unsigned long long q1 = ((dim0 >> 16) & 0xFFFFull)            // tensor_dim0[31:16]
                        | ((dim1 & 0xFFFFFFFFull) << 16)        // tensor_dim1
                        | ((tdim0 & 0xFFFFull) << 48);          // tile_dim0
  unsigned long long q2 = (tdim1 & 0xFFFFull)                   // tile_dim1 (tile_dim2=0)
                        | ((stride0 & 0xFFFFFFFFull) << 32);    // dim0_stride[31:0]
  unsigned long long q3 = (stride0 >> 32) & 0xFFFFull;          // dim0_stride[47:32]

  i32x8 g1;
  g1[0] = (int)(unsigned)(q0);  g1[1] = (int)(unsigned)(q0 >> 32);
  g1[2] = (int)(unsigned)(q1);  g1[3] = (int)(unsigned)(q1 >> 32);
  g1[4] = (int)(unsigned)(q2);  g1[5] = (int)(unsigned)(q2 >> 32);
  g1[6] = (int)(unsigned)(q3);  g1[7] = (int)(unsigned)(q3 >> 32);

  i32x4 z4 = {0, 0, 0, 0};
#if __clang_major__ >= 23
  i32x8 z8 = {0, 0, 0, 0, 0, 0, 0, 0};
  __builtin_amdgcn_tensor_load_to_lds(g0, g1, z4, z4, z8, 0);
#else
  __builtin_amdgcn_tensor_load_to_lds(g0, g1, z4, z4, 0);
#endif
}
#endif  // UPS_HAVE_TDM

__global__ __launch_bounds__(256)
void InterpolativeUpsampler_kernel(const float* __restrict__ x,
                                   float* __restrict__ out) {
  __shared__ float tile[TI + 2][WDIM];   // input rows i0-1 .. i0+TI

  const int i0    = blockIdx.x * TI;
  const int plane = blockIdx.y;          // n*C + c
  const float* src = x   + (size_t)plane * (HDIM * WDIM);
  float*       dst = out + (size_t)plane * (2 * HDIM * 2 * WDIM);

  const int t    = threadIdx.x;
  const int lane = t & 31;
  const int wv   = t >> 5;

  const int gs     = (i0 == 0) ? 0 : (i0 - 1);
  const int ge     = (i0 + TI >= HDIM) ? (HDIM - 1) : (i0 + TI);
  const int nrows  = ge - gs + 1;
  const int lstart = gs - (i0 - 1);      // 0 or 1

  // Zero-fill halo rows at the image borders (zero padding of upsampled img).
  if (i0 == 0 && t < 32) {
    f32x4 z = {0.f, 0.f, 0.f, 0.f};
    *(f32x4*)&tile[0][t * 4] = z;
  }
  if (i0 + TI >= HDIM && t < 32) {
    f32x4 z = {0.f, 0.f, 0.f, 0.f};
    *(f32x4*)&tile[TI + 1][t * 4] = z;
  }

#if UPS_HAVE_TDM
  // TDM ignores EXEC: gate with a wave-uniform scalar branch so only wave 0
  // issues the DMA, then it alone waits on TENSORcnt before the WG barrier.
  if (__builtin_amdgcn_readfirstlane(wv) == 0) {
    unsigned lds_addr = (unsigned)(size_t)(void*)&tile[lstart][0];
    tdm_load_2d(lds_addr, src + (size_t)gs * WDIM, (unsigned)nrows);
    UPS_WAIT_TENSOR();
  }
#else
  // Fallback: cooperative 16B staging.
  for (int idx = t; idx < nrows * (WDIM / 4); idx += 256) {
    int r  = idx >> 5;
    int c4 = (idx & 31) * 4;
    *(f32x4*)&tile[lstart + r][c4] =
        *(const f32x4*)&src[(size_t)(gs + r) * WDIM + c4];
  }
#endif
  __syncthreads();

  const int j0 = lane * 4;               // 4 input cols -> 8 output cols
  for (int iter = 0; iter < TI / 8; ++iter) {
    const int iloc = iter * 8 + wv;      // local input row 0..TI-1
    const int lr   = iloc + 1;           // LDS row of input row i0+iloc

    float h0[8], h1[8], h2[8];
    {
      f32x4 c  = *(const f32x4*)&tile[lr - 1][j0];
      float lf = (j0 > 0) ? tile[lr - 1][j0 - 1] : 0.f;
      float rt = (j0 + 4 < WDIM) ? tile[lr - 1][j0 + 4] : 0.f;
      hpass(lf, c, rt, h0);
    }
    {
      f32x4 c  = *(const f32x4*)&tile[lr][j0];
      float lf = (j0 > 0) ? tile[lr][j0 - 1] : 0.f;
      float rt = (j0 + 4 < WDIM) ? tile[lr][j0 + 4] : 0.f;
      hpass(lf, c, rt, h1);
    }
    {
      f32x4 c  = *(const f32x4*)&tile[lr + 1][j0];
      float lf = (j0 > 0) ? tile[lr + 1][j0 - 1] : 0.f;
      float rt = (j0 + 4 < WDIM) ? tile[lr + 1][j0 + 4] : 0.f;
      hpass(lf, c, rt, h2);
    }

    float o0[8], o1[8];
#pragma unroll
    for (int k = 0; k < 8; ++k) {
      o0[k] = 0.25f * h0[k] + 0.75f * h1[k];  // out row 2i  : rows (i-1, i)
      o1[k] = 0.75f * h1[k] + 0.25f * h2[k];  // out row 2i+1: rows (i, i+1)
    }

    const int p = 2 * (i0 + iloc);
    float* row0 = dst + (size_t)p * W2 + j0 * 2;
    float* row1 = row0 + W2;
    f32x4 s0a = {o0[0], o0[1], o0[2], o0[3]};
    f32x4 s0b = {o0[4], o0[5], o0[6], o0[7]};
    f32x4 s1a = {o1[0], o1[1], o1[2], o1[3]};
    f32x4 s1b = {o1[4], o1[5], o1[6], o1[7]};
    __builtin_nontemporal_store(s0a, (f32x4*)row0);
    __builtin_nontemporal_store(s0b, (f32x4*)(row0 + 4));
    __builtin_nontemporal_store(s1a, (f32x4*)row1);
    __builtin_nontemporal_store(s1b, (f32x4*)(row1 + 4));
  }
}

extern "C" void kernel_launch(void* const* d_in, const int* in_sizes, int n_in,
                              void* d_out, int out_size, void* d_ws, size_t ws_size,
                              hipStream_t stream) {
  (void)in_sizes; (void)n_in; (void)out_size; (void)d_ws; (void)ws_size;
  const float* x = (const float*)d_in[0];
  float* out = (float*)d_out;
  dim3 grid(HDIM / TI, 16 * 128);   // 4 row tiles x 2048 (N*C) planes
  dim3 block(256);                   // 8 wave32s
  InterpolativeUpsampler_kernel<<<grid, block, 0, stream>>>(x, out);
}